// GCN_2_hidden_model_70145405878899
// MI455X (gfx1250) — compile-verified
//
#include <hip/hip_runtime.h>
#include <hip/hip_bf16.h>

typedef __attribute__((ext_vector_type(2))) float v2f;
typedef __attribute__((ext_vector_type(8))) float v8f;

// ---------------------------------------------------------------------------
// Degree / normalization kernels
// ---------------------------------------------------------------------------
__global__ void deg_init_kernel(float* __restrict__ deg, int n) {
    int i = blockIdx.x * blockDim.x + threadIdx.x;
    if (i < n) deg[i] = 1.0f;   // self-loop contributes 1 to every node's degree
}

__global__ void deg_count_kernel(const int* __restrict__ col, float* __restrict__ deg, int E) {
    int e = blockIdx.x * blockDim.x + threadIdx.x;
    if (e < E) {
        unsafeAtomicAdd(&deg[col[e]], 1.0f);   // -> global_atomic_add_f32
    }
}

__global__ void deg_rsqrt_kernel(float* __restrict__ deg, int n) {
    int i = blockIdx.x * blockDim.x + threadIdx.x;
    if (i < n) deg[i] = rsqrtf(deg[i]);   // deg >= 1 always (self loops)
}

// ---------------------------------------------------------------------------
// Dense GEMM:  Y[M x NOUT] = act(X)[M x K] * W[K x NOUT], K/NOUT compile-time.
// One wave (32 lanes) computes one 16x16 tile via V_WMMA_F32_16X16X4_F32.
// ISA 7.12.2 fp32 layouts (wave32):
//   A 16x4 : lane L holds row M=L%16; VGPR v holds K = 2*(L>>4) + v
//   B 4x16 : lane L holds col N=L%16; VGPR v holds K = 2*(L>>4) + v
//   C/D    : lane L holds col N=L%16; VGPR r holds row M = r + 8*(L>>4)
// ---------------------------------------------------------------------------
template <bool RELU_IN, int K, int NOUT>
__global__ __launch_bounds__(32)
void gemm_wmma_f32_kernel(const float* __restrict__ X, const float* __restrict__ W,
                          float* __restrict__ Y) {
    constexpr int tilesN = NOUT / 16;
    const int lane = threadIdx.x;          // 0..31
    const int tm   = blockIdx.x / tilesN;
    const int tn   = blockIdx.x - tm * tilesN;
    const int m0   = tm << 4;
    const int n0   = tn << 4;
    const int r    = lane & 15;            // row of A / col of B,C,D
    const int half = lane >> 4;

    // Constant-stride pointers: pure pointer bumps inside the unrolled loop.
    const float* __restrict__ xp = X + (size_t)(m0 + r) * K + half * 2;
    const float* __restrict__ wp = W + (size_t)(half * 2) * NOUT + n0 + r;

    v8f c = {};
    #pragma unroll
    for (int k0 = 0; k0 < K; k0 += 4) {
        float a0 = xp[0];
        float a1 = xp[1];
        if (RELU_IN) { a0 = fmaxf(a0, 0.0f); a1 = fmaxf(a1, 0.0f); }
        v2f a = {a0, a1};
        v2f b = { wp[0], wp[NOUT] };
        // 8 args: (neg_a, A, neg_b, B, c_mod, C, reuse_a, reuse_b)
        c = __builtin_amdgcn_wmma_f32_16x16x4_f32(false, a, false, b,
                                                  (short)0, c, false, false);
        xp += 4;
        wp += 4 * NOUT;
    }

    float* __restrict__ yb = Y + (size_t)(m0 + half * 8) * NOUT + n0 + r;
    #pragma unroll
    for (int i = 0; i < 8; ++i)
        yb[i * NOUT] = c[i];
}

// ---------------------------------------------------------------------------
// Aggregation epilogue init: out[i][j] = bias[j] + H[i][j] * dinv[i]^2
// (bias + self-loop message). Also initializes the accumulation target.
// ---------------------------------------------------------------------------
__global__ void agg_init_kernel(const float* __restrict__ H, const float* __restrict__ dinv,
                                const float* __restrict__ bias, float* __restrict__ out,
                                int n, int d) {
    size_t i = (size_t)blockIdx.x * blockDim.x + threadIdx.x;
    size_t total = (size_t)n * d;
    if (i < total) {
        int node = (int)(i / d);
        int j    = (int)(i - (size_t)node * d);
        float w  = dinv[node];
        out[i] = bias[j] + H[i] * (w * w);
    }
}

// ---------------------------------------------------------------------------
// Edge scatter: out[col] += H[row] * (dinv[row]*dinv[col])
// (d>>2) threads per edge, float4 gather + 4 fp32 hardware atomic adds.
// dlog2 = log2(d) (7 for 128, 6 for 64).
// ---------------------------------------------------------------------------
__global__ void agg_scatter_kernel(const float* __restrict__ H,
                                   const int* __restrict__ row,
                                   const int* __restrict__ col,
                                   const float* __restrict__ dinv,
                                   float* __restrict__ out,
                                   int E, int dlog2) {
    const int tpe_shift = dlog2 - 2;                       // threads per edge = d/4
    size_t gid = (size_t)blockIdx.x * blockDim.x + threadIdx.x;
    size_t e   = gid >> tpe_shift;
    if (e >= (size_t)E) return;
    int j = (int)(gid & (((size_t)1 << tpe_shift) - 1)) << 2;   // float offset
    int rn = row[e];
    int cn = col[e];
    float norm = dinv[rn] * dinv[cn];
    const float4 h = *(const float4*)(H + ((size_t)rn << dlog2) + j);
    float* o = out + ((size_t)cn << dlog2) + j;
    unsafeAtomicAdd(o + 0, h.x * norm);
    unsafeAtomicAdd(o + 1, h.y * norm);
    unsafeAtomicAdd(o + 2, h.z * norm);
    unsafeAtomicAdd(o + 3, h.w * norm);
}

// ---------------------------------------------------------------------------
// Host-side orchestration
// ---------------------------------------------------------------------------
extern "C" void kernel_launch(void* const* d_in, const int* in_sizes, int n_in,
                              void* d_out, int out_size, void* d_ws, size_t ws_size,
                              hipStream_t stream) {
    const float* x   = (const float*)d_in[0];
    const int*   ei  = (const int*)d_in[1];
    const float* W1  = (const float*)d_in[2];
    const float* b1  = (const float*)d_in[3];
    const float* W2  = (const float*)d_in[4];
    const float* b2  = (const float*)d_in[5];
    const float* W3  = (const float*)d_in[6];
    const float* b3  = (const float*)d_in[7];

    const int IN_DIM = 128, HID = 128;
    const int N = in_sizes[0] / IN_DIM;        // 40000
    const int E = in_sizes[1] / 2;             // 640000
    const int OUT_DIM = out_size / N;          // 64
    const int* row = ei;                       // edge_index[0]
    const int* col = ei + E;                   // edge_index[1]

    // Workspace layout: dinv[N] | G[N*128] | P[N*128]  (~41 MB, L2-resident)
    float* dinv = (float*)d_ws;
    float* G    = dinv + N;
    float* P    = G + (size_t)N * HID;

    float* out = (float*)d_out;

    const int T = 256;

    // --- degree -> dinv ---
    deg_init_kernel<<<(N + T - 1) / T, T, 0, stream>>>(dinv, N);
    deg_count_kernel<<<(E + T - 1) / T, T, 0, stream>>>(col, dinv, E);
    deg_rsqrt_kernel<<<(N + T - 1) / T, T, 0, stream>>>(dinv, N);

    const int tilesM     = N / 16;                       // 2500
    const size_t nd128   = (size_t)N * HID;
    const size_t ndOut   = (size_t)N * OUT_DIM;
    const size_t scat128 = (size_t)E * (HID >> 2);       // threads for d=128 scatter
    const size_t scatOut = (size_t)E * (OUT_DIM >> 2);

    // --- layer 1: G = x @ W1 ; P = agg(G) + b1 (relu deferred to next gemm) ---
    gemm_wmma_f32_kernel<false, 128, 128><<<tilesM * (HID / 16), 32, 0, stream>>>(x, W1, G);
    agg_init_kernel<<<(int)((nd128 + T - 1) / T), T, 0, stream>>>(G, dinv, b1, P, N, HID);
    agg_scatter_kernel<<<(int)((scat128 + T - 1) / T), T, 0, stream>>>(G, row, col, dinv, P, E, 7);

    // --- layer 2: G = relu(P) @ W2 ; P = agg(G) + b2 ---
    gemm_wmma_f32_kernel<true, 128, 128><<<tilesM * (HID / 16), 32, 0, stream>>>(P, W2, G);
    agg_init_kernel<<<(int)((nd128 + T - 1) / T), T, 0, stream>>>(G, dinv, b2, P, N, HID);
    agg_scatter_kernel<<<(int)((scat128 + T - 1) / T), T, 0, stream>>>(G, row, col, dinv, P, E, 7);

    // --- layer 3: G = relu(P) @ W3 ; out = agg(G) + b3 (no relu) ---
    gemm_wmma_f32_kernel<true, 128, 64><<<tilesM * (OUT_DIM / 16), 32, 0, stream>>>(P, W3, G);
    agg_init_kernel<<<(int)((ndOut + T - 1) / T), T, 0, stream>>>(G, dinv, b3, out, N, OUT_DIM);
    agg_scatter_kernel<<<(int)((scatOut + T - 1) / T), T, 0, stream>>>(G, row, col, dinv, out, E, 6);
}